// StructuralAttention_10179072491849
// MI455X (gfx1250) — compile-verified
//
#include <hip/hip_runtime.h>
#include <hip/hip_bf16.h>

typedef __attribute__((ext_vector_type(2))) float v2f;
typedef __attribute__((ext_vector_type(8))) float v8f;

// Branch-free tanh: prefer gfx1250 hardware V_TANH_F32 (single TRANS op).
static __device__ __forceinline__ float fast_tanh(float x) {
#if __has_builtin(__builtin_amdgcn_tanhf)
    return __builtin_amdgcn_tanhf(x);
#elif __has_builtin(__builtin_amdgcn_tanh_f32)
    return __builtin_amdgcn_tanh_f32(x);
#else
    // tanh(x) = 1 - 2/(exp(2x)+1); saturates correctly at +/-1 for large |x|.
    const float e = __expf(2.0f * x);
    return 1.0f - 2.0f / (e + 1.0f);
#endif
}

// ---------------------------------------------------------------------------
// Phase 1: raw[j] = exp( dot(tanh(x[j,:]), w) ), val[n-1] = 1.0
// One wave32 computes 16 node scores via V_WMMA_F32_16X16X4_F32 (fp32 exact).
// A (16x4 f32): lane L<16 -> M=L, K=0/1 in v0/v1 ; lane L>=16 -> M=L-16, K=2/3.
// B (4x16 f32): w chunk replicated across all 16 columns, so every column of
// D holds the dot product. K-loop unrolled x4: 8 b64 loads in flight per iter.
// ---------------------------------------------------------------------------
__global__ void sa_scores_wmma(const float* __restrict__ x,
                               const float* __restrict__ w,
                               float* __restrict__ val, int n, int d) {
    const int lane = threadIdx.x & 31;
    const int wave = blockIdx.x * (blockDim.x >> 5) + (threadIdx.x >> 5);
    const int base = wave * 16;
    if (base >= n) return;  // uniform per wave; EXEC stays all-ones for WMMA

    int row = base + (lane & 15);
    if (row >= n) row = n - 1;  // clamp (result discarded by store guard)
    const float* xr = x + (size_t)row * d;
    const int off = (lane < 16) ? 0 : 2;

    v8f acc = {0.f, 0.f, 0.f, 0.f, 0.f, 0.f, 0.f, 0.f};
    for (int c = 0; c < d; c += 16) {
        v2f a[4], b[4];
#pragma unroll
        for (int u = 0; u < 4; ++u) {
            a[u] = *(const v2f*)(xr + c + 4 * u + off);  // 8B aligned
            b[u] = *(const v2f*)(w  + c + 4 * u + off);
        }
#pragma unroll
        for (int u = 0; u < 4; ++u) {
            v2f ta;
            ta.x = fast_tanh(a[u].x);
            ta.y = fast_tanh(a[u].y);
            // 8 args: (neg_a, A, neg_b, B, c_mod, C, reuse_a, reuse_b)
            acc = __builtin_amdgcn_wmma_f32_16x16x4_f32(
                false, ta, false, b[u], (short)0, acc, false, false);
        }
    }

    // Column 0 of D lives in lanes 0 (M=0..7) and 16 (M=8..15).
    if ((lane & 15) == 0) {
        const int mbase = base + ((lane >> 4) << 3);
#pragma unroll
        for (int t = 0; t < 8; ++t) {
            const int node = mbase + t;
            if (node < n)
                val[node] = (node == n - 1) ? 1.0f : expf(acc[t]);
        }
    }
}

// ---------------------------------------------------------------------------
// Phase 2: top-down renormalization. val (256 KB) is L2-resident; a single
// 1024-thread block walks the 15 internal levels with fence+barrier between
// levels. Level d: parents j in [n+1-2^(d+1), n-2^d], children disjoint.
// ---------------------------------------------------------------------------
__global__ void sa_topdown(float* __restrict__ val,
                           const int* __restrict__ left,
                           const int* __restrict__ right,
                           int n, int levels) {
    for (int d = 0; d < levels; ++d) {
        const int cnt = 1 << d;
        const int lo = n + 1 - (cnt << 1);
        for (int t = threadIdx.x; t < cnt; t += blockDim.x) {
            const int j = lo + t;
            const int l = left[j];
            const int r = right[j];
            const float vi = val[j];
            const float m = vi / (val[l] + val[r] + 1e-10f);
            val[l] *= m;
            val[r] *= m;
        }
        __threadfence();
        __syncthreads();
    }
}

// ---------------------------------------------------------------------------
// Phase 3a: leaves (rows [0, (n+1)/2)) pass through c0/h0 unchanged.
// ---------------------------------------------------------------------------
__global__ void sa_leafcopy(const float* __restrict__ c0,
                            const float* __restrict__ h0,
                            float* __restrict__ outC,
                            float* __restrict__ outH, int total) {
    const int i = blockIdx.x * blockDim.x + threadIdx.x;
    if (i < total) {
        outC[i] = c0[i];
        outH[i] = h0[i];
    }
}

// ---------------------------------------------------------------------------
// Phase 3b: one launch per internal level (deepest first):
//   C[p,:] = C[l,:]*val[l] + C[r,:]*val[r]   (same for H)
// Fully coalesced: 256 consecutive threads cover one parent row.
// ---------------------------------------------------------------------------
__global__ void sa_combine(float* __restrict__ outC,
                           float* __restrict__ outH,
                           const float* __restrict__ val,
                           const int* __restrict__ left,
                           const int* __restrict__ right,
                           int lo, int cnt, int d) {
    const int tid = blockIdx.x * blockDim.x + threadIdx.x;
    const int total = cnt * d;
    if (tid >= total) return;
    const int p = lo + tid / d;
    const int col = tid % d;
    const int l = left[p];
    const int r = right[p];
    const float vl = val[l];
    const float vr = val[r];
    const size_t pi = (size_t)p * d + col;
    const size_t li = (size_t)l * d + col;
    const size_t ri = (size_t)r * d + col;
    outC[pi] = fmaf(outC[li], vl, outC[ri] * vr);
    outH[pi] = fmaf(outH[li], vl, outH[ri] * vr);
}

extern "C" void kernel_launch(void* const* d_in, const int* in_sizes, int n_in,
                              void* d_out, int out_size, void* d_ws, size_t ws_size,
                              hipStream_t stream) {
    (void)n_in; (void)out_size; (void)d_ws; (void)ws_size;

    const float* x    = (const float*)d_in[0];
    const float* w    = (const float*)d_in[1];
    const float* c0   = (const float*)d_in[2];
    const float* h0   = (const float*)d_in[3];
    const int*   left = (const int*)d_in[4];
    const int*   right= (const int*)d_in[5];

    const int n = in_sizes[4];   // N = 65535
    const int d = in_sizes[1];   // D = 256

    float* val  = (float*)d_out;          // [n]
    float* outC = val + n;                // [n, d]
    float* outH = outC + (size_t)n * d;   // [n, d]

    int depth = 0;
    while ((1 << depth) < n + 1) depth++;  // 16 for N = 2^16 - 1
    const int internalLevels = depth - 1;  // 15

    // 1) scores: 16 nodes per wave, 8 waves per block
    {
        const int waves  = (n + 15) / 16;
        const int blocks = (waves + 7) / 8;
        sa_scores_wmma<<<blocks, 256, 0, stream>>>(x, w, val, n, d);
    }

    // 2) top-down scan: single block, barrier per level
    sa_topdown<<<1, 1024, 0, stream>>>(val, left, right, n, internalLevels);

    // 3a) leaf passthrough
    {
        const int nleaf = (n + 1) / 2;
        const int total = nleaf * d;
        sa_leafcopy<<<(total + 255) / 256, 256, 0, stream>>>(c0, h0, outC, outH, total);
    }

    // 3b) bottom-up combine, deepest internal level first
    for (int dd = internalLevels - 1; dd >= 0; --dd) {
        const int cnt = 1 << dd;
        const int lo  = n + 1 - (cnt << 1);
        const int tot = cnt * d;
        sa_combine<<<(tot + 255) / 256, 256, 0, stream>>>(
            outC, outH, val, left, right, lo, cnt, d);
    }
}